// GNN15_27410481283384
// MI455X (gfx1250) — compile-verified
//
#include <hip/hip_runtime.h>
#include <math.h>

// GNN15: two GAT branches + node-softmax attention pooling, fp32 throughout.
// WMMA: V_WMMA_F32_16X16X4_F32 for the fused node GEMM
//   [16x12] x [12x64] where cols 0-47 = W, 48-50 = W@a_src, 51-53 = W@a_dst.
// B staged in LDS via GLOBAL_LOAD_ASYNC_TO_LDS (ASYNCcnt) when available.
// Edge passes: one thread per edge (3 heads in-thread), int32-converted indices.
// Needs ~104 MB of d_ws.

#define HEADS 3
#define FILT 16
#define VF 11
#define NN 100000
#define EI 400000
#define ENH 1600000
#define NTILES (NN / 16)

typedef __attribute__((ext_vector_type(2))) float v2f;
typedef __attribute__((ext_vector_type(8))) float v8f;

// ---- workspace layout (float offsets) ----
constexpr size_t OFF_BPAD   = 0;                              // 2*12*64
constexpr size_t OFF_H_INT  = OFF_BPAD + 2 * 12 * 64;         // NN*48
constexpr size_t OFF_H_NH   = OFF_H_INT + (size_t)NN * 48;    // NN*48
constexpr size_t OFF_ES_INT = OFF_H_NH + (size_t)NN * 48;     // NN*3
constexpr size_t OFF_ED_INT = OFF_ES_INT + (size_t)NN * 3;
constexpr size_t OFF_ES_NH  = OFF_ED_INT + (size_t)NN * 3;
constexpr size_t OFF_ED_NH  = OFF_ES_NH + (size_t)NN * 3;
constexpr size_t OFF_SCORES = OFF_ED_NH + (size_t)NN * 3;     // NN*3
// --- zero-init block ---
constexpr size_t OFF_X      = OFF_SCORES + (size_t)NN * 3;    // NN*96 (concat accum)
constexpr size_t OFF_DENI   = OFF_X + (size_t)NN * 96;        // NN*3
constexpr size_t OFF_DENN   = OFF_DENI + (size_t)NN * 3;      // NN*3
constexpr size_t OFF_ESUM   = OFF_DENN + (size_t)NN * 3;      // 4
constexpr size_t OFF_REP    = OFF_ESUM + 4;                   // 288
constexpr size_t ZERO_BEGIN = OFF_X;
constexpr size_t ZERO_END   = OFF_REP + 288;
// --- (-inf)-init block ---
constexpr size_t OFF_MI     = ZERO_END;                       // NN*3
constexpr size_t OFF_MN     = OFF_MI + (size_t)NN * 3;        // NN*3
constexpr size_t OFF_SMAX   = OFF_MN + (size_t)NN * 3;        // 4
constexpr size_t NEG_BEGIN  = OFF_MI;
constexpr size_t NEG_END    = OFF_SMAX + 4;
// --- int32 index copies (stored in ws, reinterpreted as int) ---
constexpr size_t OFF_IDXI   = NEG_END;                        // 2*EI ints
constexpr size_t OFF_IDXN   = OFF_IDXI + (size_t)2 * EI;      // 2*ENH ints

// order-preserving float atomic max (targets initialized to -inf, no NaNs)
__device__ __forceinline__ void atomicMaxFloat(float* addr, float v) {
  int bits = __float_as_int(v);
  if (bits >= 0) atomicMax((int*)addr, bits);
  else           atomicMin((unsigned int*)addr, (unsigned int)bits);
}

__global__ void k_fill(float* __restrict__ p, float v, int n) {
  for (int i = blockIdx.x * blockDim.x + threadIdx.x; i < n; i += gridDim.x * blockDim.x)
    p[i] = v;
}

// int64 -> int32 edge index conversion (indices < NN fit in i32)
__global__ void k_cvt(const long long* __restrict__ src, int* __restrict__ dst, int n) {
  int i = blockIdx.x * blockDim.x + threadIdx.x;
  if (i < n) dst[i] = (int)src[i];
}

// Build padded B matrices: [2 branches][12 k][64 cols]
__global__ void k_bprep(const float* __restrict__ Wi, const float* __restrict__ asi,
                        const float* __restrict__ adi, const float* __restrict__ Wn,
                        const float* __restrict__ asn, const float* __restrict__ adn,
                        float* __restrict__ bpad) {
  int i = blockIdx.x * blockDim.x + threadIdx.x;
  if (i >= 2 * 12 * 64) return;
  int br  = i / (12 * 64);
  int k   = (i / 64) % 12;
  int col = i % 64;
  const float* W  = br ? Wn  : Wi;
  const float* as = br ? asn : asi;
  const float* ad = br ? adn : adi;
  float v = 0.0f;
  if (k < VF) {
    if (col < 48) {
      v = W[k * 48 + col];
    } else if (col < 51) {
      int h = col - 48; float acc = 0.0f;
      for (int f = 0; f < FILT; ++f) acc += W[k * 48 + h * FILT + f] * as[h * FILT + f];
      v = acc;
    } else if (col < 54) {
      int h = col - 51; float acc = 0.0f;
      for (int f = 0; f < FILT; ++f) acc += W[k * 48 + h * FILT + f] * ad[h * FILT + f];
      v = acc;
    }
  }
  bpad[i] = v;
}

// 8 waves per block; B staged once per block into LDS (async-to-LDS when
// available), then each wave runs 24x V_WMMA_F32_16X16X4_F32 on its 16-node tile.
__global__ void __launch_bounds__(256)
k_gemm(const float* __restrict__ x, const float* __restrict__ bpad,
       float* __restrict__ h0, float* __restrict__ h1,
       float* __restrict__ es0, float* __restrict__ ed0,
       float* __restrict__ es1, float* __restrict__ ed1) {
  __shared__ float bsh[2 * 12 * 64];

  // ---- stage B into LDS ----
#if __has_builtin(__builtin_amdgcn_global_load_async_to_lds_b32)
  {
    typedef __attribute__((address_space(3))) int lds_i;
    typedef __attribute__((address_space(1))) int gl_i;
    lds_i* lp = (lds_i*)bsh;
    gl_i*  gp = (gl_i*)bpad;
    for (int i = threadIdx.x; i < 2 * 12 * 64; i += blockDim.x)
      __builtin_amdgcn_global_load_async_to_lds_b32(gp + i, lp + i, 0, 0);
#if __has_builtin(__builtin_amdgcn_s_wait_asynccnt)
    __builtin_amdgcn_s_wait_asynccnt(0);
#else
    asm volatile("s_wait_asynccnt 0x0" ::: "memory");
#endif
  }
#else
  for (int i = threadIdx.x; i < 2 * 12 * 64; i += blockDim.x) bsh[i] = bpad[i];
#endif
  __syncthreads();   // before any wave-level divergence

  const int wave  = threadIdx.x >> 5;
  const int tile  = blockIdx.x * 8 + wave;
  if (tile >= NTILES) return;                 // uniform per wave: EXEC stays full
  const int lane  = threadIdx.x & 31;
  const int lrow  = lane & 15;
  const int hi    = lane >> 4;                // 0 or 1 (lane half)
  const int khalf = hi * 2;

  // A fragments: 16x4 f32 per k-step; lane holds (row = lane%16, K = khalf + j)
  v2f a[3];
#pragma unroll
  for (int ks = 0; ks < 3; ++ks) {
#pragma unroll
    for (int j = 0; j < 2; ++j) {
      int k = ks * 4 + khalf + j;
      a[ks][j] = (k < VF) ? x[(size_t)(tile * 16 + lrow) * VF + k] : 0.0f;
    }
  }

  for (int br = 0; br < 2; ++br) {
    const float* B  = bsh + (size_t)br * 12 * 64;
    float* hbuf = br ? h1 : h0;
    float* es   = br ? es1 : es0;
    float* ed   = br ? ed1 : ed0;
#pragma unroll
    for (int nt = 0; nt < 4; ++nt) {
      v8f acc = {};
      const int col = nt * 16 + lrow;
#pragma unroll
      for (int ks = 0; ks < 3; ++ks) {
        v2f b;
        b[0] = B[(ks * 4 + khalf + 0) * 64 + col];
        b[1] = B[(ks * 4 + khalf + 1) * 64 + col];
        acc = __builtin_amdgcn_wmma_f32_16x16x4_f32(
            /*neg_a=*/false, a[ks], /*neg_b=*/false, b,
            /*c_mod=*/(short)0, acc, /*reuse_a=*/false, /*reuse_b=*/false);
      }
      // D layout: VGPR v -> row v + 8*hi, col = nt*16 + lane%16
      const int rbase = tile * 16 + hi * 8;
#pragma unroll
      for (int v = 0; v < 8; ++v) {
        const int r = rbase + v;
        const float val = acc[v];
        if (nt < 3) {
          hbuf[(size_t)r * 48 + col] = val;
        } else {
          const int c = col - 48;               // 0..15
          if (c < 3)      es[(size_t)r * 3 + c] = val;
          else if (c < 6) ed[(size_t)r * 3 + (c - 3)] = val;
        }
      }
    }
  }
}

// ---- edge passes: one thread per edge, 3 heads in-thread ----
__global__ void k_edge_max(const int* __restrict__ idx, int E,
                           const float* __restrict__ es, const float* __restrict__ ed,
                           float* __restrict__ m) {
  int e = blockIdx.x * blockDim.x + threadIdx.x;
  if (e >= E) return;
  int s = idx[e], d = idx[E + e];
#pragma unroll
  for (int h = 0; h < HEADS; ++h) {
    float v = es[s * 3 + h] + ed[d * 3 + h];
    v = v > 0.0f ? v : 0.2f * v;               // leaky_relu(0.2)
    atomicMaxFloat(&m[d * 3 + h], v);
  }
}

__global__ void k_edge_sum(const int* __restrict__ idx, int E,
                           const float* __restrict__ es, const float* __restrict__ ed,
                           const float* __restrict__ m, float* __restrict__ den) {
  int e = blockIdx.x * blockDim.x + threadIdx.x;
  if (e >= E) return;
  int s = idx[e], d = idx[E + e];
#pragma unroll
  for (int h = 0; h < HEADS; ++h) {
    float v = es[s * 3 + h] + ed[d * 3 + h];
    v = v > 0.0f ? v : 0.2f * v;
    atomicAdd(&den[d * 3 + h], __expf(v - m[d * 3 + h]));
  }
}

__global__ void k_edge_scatter(const int* __restrict__ idx, int E,
                               const float* __restrict__ es, const float* __restrict__ ed,
                               const float* __restrict__ m, const float* __restrict__ den,
                               const float* __restrict__ hbuf, float* __restrict__ x,
                               int colOff) {
  int e = blockIdx.x * blockDim.x + threadIdx.x;
  if (e >= E) return;
  int s = idx[e], d = idx[E + e];
  const float* hs = hbuf + (size_t)s * 48;
  float* xd = x + (size_t)d * 96 + colOff;
#pragma unroll
  for (int h = 0; h < HEADS; ++h) {
    float v = es[s * 3 + h] + ed[d * 3 + h];
    v = v > 0.0f ? v : 0.2f * v;
    float p = __expf(v - m[d * 3 + h]);
    float alpha = p / (den[d * 3 + h] + 1e-16f);
#pragma unroll
    for (int f = 0; f < FILT; ++f)
      atomicAdd(&xd[h * FILT + f], alpha * hs[h * FILT + f]);
  }
}

__global__ void k_scores(const float* __restrict__ x, const float* __restrict__ attw,
                         float* __restrict__ scores, float* __restrict__ smax) {
  int n = blockIdx.x * blockDim.x + threadIdx.x;
  if (n >= NN) return;
  const float* xr = x + (size_t)n * 96;
  float s0 = 0, s1 = 0, s2 = 0;
  for (int d = 0; d < 96; ++d) {
    float xv = xr[d];
    s0 += xv * attw[d * 3 + 0];
    s1 += xv * attw[d * 3 + 1];
    s2 += xv * attw[d * 3 + 2];
  }
  s0 = tanhf(s0); s1 = tanhf(s1); s2 = tanhf(s2);
  scores[n * 3 + 0] = s0; scores[n * 3 + 1] = s1; scores[n * 3 + 2] = s2;
  atomicMaxFloat(&smax[0], s0);
  atomicMaxFloat(&smax[1], s1);
  atomicMaxFloat(&smax[2], s2);
}

__global__ void k_pool(const float* __restrict__ x, const float* __restrict__ scores,
                       const float* __restrict__ smax, float* __restrict__ esum,
                       float* __restrict__ rep) {
  __shared__ float lrep[288];
  __shared__ float lesum[3];
  for (int i = threadIdx.x; i < 288; i += blockDim.x) lrep[i] = 0.0f;
  if (threadIdx.x < 3) lesum[threadIdx.x] = 0.0f;
  __syncthreads();
  for (int n = blockIdx.x * blockDim.x + threadIdx.x; n < NN; n += gridDim.x * blockDim.x) {
    float p0 = __expf(scores[n * 3 + 0] - smax[0]);
    float p1 = __expf(scores[n * 3 + 1] - smax[1]);
    float p2 = __expf(scores[n * 3 + 2] - smax[2]);
    atomicAdd(&lesum[0], p0); atomicAdd(&lesum[1], p1); atomicAdd(&lesum[2], p2);
    const float* xr = x + (size_t)n * 96;
    for (int d = 0; d < 96; ++d) {
      float xv = xr[d];
      atomicAdd(&lrep[0 * 96 + d], p0 * xv);
      atomicAdd(&lrep[1 * 96 + d], p1 * xv);
      atomicAdd(&lrep[2 * 96 + d], p2 * xv);
    }
  }
  __syncthreads();
  for (int i = threadIdx.x; i < 288; i += blockDim.x) atomicAdd(&rep[i], lrep[i]);
  if (threadIdx.x < 3) atomicAdd(&esum[threadIdx.x], lesum[threadIdx.x]);
}

__global__ void k_final(const float* __restrict__ rep, const float* __restrict__ esum,
                        const float* __restrict__ dw, const float* __restrict__ db,
                        float* __restrict__ out) {
  if (blockIdx.x == 0 && threadIdx.x == 0) {
    float acc = db[0];
    for (int h = 0; h < HEADS; ++h) {
      float inv = 1.0f / esum[h];
      for (int d = 0; d < 96; ++d) acc += rep[h * 96 + d] * inv * dw[h * 96 + d];
    }
    out[0] = acc;
  }
}

extern "C" void kernel_launch(void* const* d_in, const int* in_sizes, int n_in,
                              void* d_out, int out_size, void* d_ws, size_t ws_size,
                              hipStream_t stream) {
  const float* node_feats = (const float*)d_in[0];
  const float* W_int      = (const float*)d_in[1];
  const float* a_src_int  = (const float*)d_in[2];
  const float* a_dst_int  = (const float*)d_in[3];
  const float* W_nh       = (const float*)d_in[4];
  const float* a_src_nh   = (const float*)d_in[5];
  const float* a_dst_nh   = (const float*)d_in[6];
  const float* att_w      = (const float*)d_in[7];
  const float* dense_w    = (const float*)d_in[8];
  const float* dense_b    = (const float*)d_in[9];
  const long long* edge_int = (const long long*)d_in[10];  // jnp.int64
  const long long* edge_nh  = (const long long*)d_in[11];

  float* ws = (float*)d_ws;
  float* bpad   = ws + OFF_BPAD;
  float* h_int  = ws + OFF_H_INT;
  float* h_nh   = ws + OFF_H_NH;
  float* es_int = ws + OFF_ES_INT;
  float* ed_int = ws + OFF_ED_INT;
  float* es_nh  = ws + OFF_ES_NH;
  float* ed_nh  = ws + OFF_ED_NH;
  float* scores = ws + OFF_SCORES;
  float* x      = ws + OFF_X;
  float* den_i  = ws + OFF_DENI;
  float* den_n  = ws + OFF_DENN;
  float* esum   = ws + OFF_ESUM;
  float* rep    = ws + OFF_REP;
  float* m_i    = ws + OFF_MI;
  float* m_n    = ws + OFF_MN;
  float* smax   = ws + OFF_SMAX;
  int*   idx_i  = (int*)(ws + OFF_IDXI);
  int*   idx_n  = (int*)(ws + OFF_IDXN);

  // init accumulators
  {
    int zn = (int)(ZERO_END - ZERO_BEGIN);
    k_fill<<<(zn + 1023) / 1024, 256, 0, stream>>>(ws + ZERO_BEGIN, 0.0f, zn);
    int nn = (int)(NEG_END - NEG_BEGIN);
    k_fill<<<(nn + 1023) / 1024, 256, 0, stream>>>(ws + NEG_BEGIN, -INFINITY, nn);
  }

  // int64 -> int32 index conversion (halves index bandwidth for 3 passes)
  k_cvt<<<(2 * EI + 255) / 256, 256, 0, stream>>>(edge_int, idx_i, 2 * EI);
  k_cvt<<<(2 * ENH + 255) / 256, 256, 0, stream>>>(edge_nh, idx_n, 2 * ENH);

  // padded B (W | W@a_src | W@a_dst) for both branches
  k_bprep<<<(2 * 12 * 64 + 255) / 256, 256, 0, stream>>>(
      W_int, a_src_int, a_dst_int, W_nh, a_src_nh, a_dst_nh, bpad);

  // fused WMMA GEMM: h, e_src, e_dst for both branches
  k_gemm<<<(NTILES + 7) / 8, 256, 0, stream>>>(node_feats, bpad, h_int, h_nh,
                                               es_int, ed_int, es_nh, ed_nh);

  // branch "int"
  {
    int b = (EI + 255) / 256;
    k_edge_max<<<b, 256, 0, stream>>>(idx_i, EI, es_int, ed_int, m_i);
    k_edge_sum<<<b, 256, 0, stream>>>(idx_i, EI, es_int, ed_int, m_i, den_i);
    k_edge_scatter<<<b, 256, 0, stream>>>(idx_i, EI, es_int, ed_int, m_i, den_i,
                                          h_int, x, 0);
  }
  // branch "nh"
  {
    int b = (ENH + 255) / 256;
    k_edge_max<<<b, 256, 0, stream>>>(idx_n, ENH, es_nh, ed_nh, m_n);
    k_edge_sum<<<b, 256, 0, stream>>>(idx_n, ENH, es_nh, ed_nh, m_n, den_n);
    k_edge_scatter<<<b, 256, 0, stream>>>(idx_n, ENH, es_nh, ed_nh, m_n, den_n,
                                          h_nh, x, 48);
  }

  // pooling
  k_scores<<<(NN + 255) / 256, 256, 0, stream>>>(x, att_w, scores, smax);
  k_pool<<<400, 256, 0, stream>>>(x, scores, smax, esum, rep);
  k_final<<<1, 64, 0, stream>>>(rep, esum, dense_w, dense_b, (float*)d_out);
}